// DetectionLoss_73882027426142
// MI455X (gfx1250) — compile-verified
//
#include <hip/hip_runtime.h>
#include <hip/hip_bf16.h>
#include <math.h>

typedef __attribute__((ext_vector_type(2))) float v2f;
typedef __attribute__((ext_vector_type(8))) float v8f;

#define BOX_W 5.0f
#define GIOU_W 2.0f
#define OBJ_W 1.0f
#define CLS_W 1.5f
#define CTR_W 0.5f

__device__ __forceinline__ float bce_logits(float x, float t) {
    return fmaxf(x, 0.0f) - x * t + log1pf(expf(-fabsf(x)));
}

__device__ __forceinline__ float focal_elem(float x, float t) {
    float p  = 1.0f / (1.0f + expf(-x));
    float ce = bce_logits(x, t);
    float pt = p * t + (1.0f - p) * (1.0f - t);
    float at = 0.25f * t + 0.75f * (1.0f - t);
    float om = 1.0f - pt;
    return at * ce * om * om;
}

// ---------------------------------------------------------------------------
// Pass A: per (b,m) GT box, reduce over all L locations:
//   has_candidate[bm] = any(inside_box & inside_center)
//   best[bm] = argmin(dist | inside_box) if any inside_box else argmin(dist)
// jnp.argmin tie-break = lowest index; replicated via (val, idx) combine.
// ---------------------------------------------------------------------------
__global__ __launch_bounds__(256)
void fcos_assign(const float* __restrict__ loc,
                 const float* __restrict__ gt_boxes,
                 const int* __restrict__ gh, const int* __restrict__ gw,
                 int L,
                 int* __restrict__ best, int* __restrict__ hc) {
    const int bm  = blockIdx.x;
    const int tid = threadIdx.x;
    const float cx = gt_boxes[bm * 4 + 0], cy = gt_boxes[bm * 4 + 1];
    const float w  = gt_boxes[bm * 4 + 2], h  = gt_boxes[bm * 4 + 3];
    const float x1 = cx - 0.5f * w, y1 = cy - 0.5f * h;
    const float x2 = cx + 0.5f * w, y2 = cy + 0.5f * h;
    const float rx = 1.0f / (float)gw[0], ry = 1.0f / (float)gh[0];

    float vin = __builtin_inff(); int iin = 0x7fffffff;
    float val = __builtin_inff(); int ial = 0x7fffffff;
    int flags = 0;

    for (int l = tid; l < L; l += blockDim.x) {
        float lx = loc[2 * l], ly = loc[2 * l + 1];
        float dx = lx - cx, dy = ly - cy;
        float dist = dx * dx + dy * dy;
        bool ib = (lx > x1) && (ly > y1) && (lx < x2) && (ly < y2);
        bool ic = (fabsf(dx) <= rx) && (fabsf(dy) <= ry);
        if (dist < val) { val = dist; ial = l; }
        if (ib) {
            flags |= 1;
            if (dist < vin) { vin = dist; iin = l; }
            if (ic) flags |= 2;
        }
    }

    __shared__ float s_vin[256]; __shared__ int s_iin[256];
    __shared__ float s_val[256]; __shared__ int s_ial[256];
    __shared__ int   s_fl[256];
    s_vin[tid] = vin; s_iin[tid] = iin;
    s_val[tid] = val; s_ial[tid] = ial;
    s_fl[tid] = flags;
    __syncthreads();
    for (int s = blockDim.x >> 1; s > 0; s >>= 1) {
        if (tid < s) {
            float v2 = s_vin[tid + s]; int i2 = s_iin[tid + s];
            if (v2 < s_vin[tid] || (v2 == s_vin[tid] && i2 < s_iin[tid])) { s_vin[tid] = v2; s_iin[tid] = i2; }
            v2 = s_val[tid + s]; i2 = s_ial[tid + s];
            if (v2 < s_val[tid] || (v2 == s_val[tid] && i2 < s_ial[tid])) { s_val[tid] = v2; s_ial[tid] = i2; }
            s_fl[tid] |= s_fl[tid + s];
        }
        __syncthreads();
    }
    if (tid == 0) {
        int f = s_fl[0];
        best[bm] = (f & 1) ? s_iin[0] : s_ial[0];
        hc[bm]   = (f & 2) ? 1 : 0;
    }
}

// ---------------------------------------------------------------------------
// Main kernel: one location per thread; 256 contiguous locations per block.
// Phase 1: assignment + obj focal + smooth-L1 + GIoU + centerness BCE.
// Phase 2: cooperative, fully-coalesced class-focal sweep over 256*C logits.
// Deterministic shuffle + LDS reduction -> 6 partials per block.
// ---------------------------------------------------------------------------
__global__ __launch_bounds__(256)
void fcos_main(const float* __restrict__ boxes_xyxy,
               const float* __restrict__ box_deltas,
               const float* __restrict__ class_logits,
               const float* __restrict__ objectness,
               const float* __restrict__ centerness,
               const float* __restrict__ loc,
               const float* __restrict__ gt_boxes,
               const int* __restrict__ gt_labels,
               const int* __restrict__ gh, const int* __restrict__ gw,
               const int* __restrict__ best, const int* __restrict__ hc,
               int M, int L, int C,
               float* __restrict__ partials) {
    const int tid = threadIdx.x;
    const int blocksPerBatch = L / 256;
    const int b  = blockIdx.x / blocksPerBatch;
    const int l0 = (blockIdx.x % blocksPerBatch) * 256;
    const int l  = l0 + tid;

    __shared__ float s_box[64][4];
    __shared__ int   s_glab[64];
    __shared__ int   s_best[64];
    __shared__ int   s_hc[64];
    __shared__ int   s_lab[256];
    __shared__ float s_red[8][6];

    if (tid < M) {
        s_glab[tid] = gt_labels[b * M + tid];
        s_best[tid] = best[b * M + tid];
        s_hc[tid]   = hc[b * M + tid];
    }
    if (tid < M * 4) s_box[tid >> 2][tid & 3] = gt_boxes[b * M * 4 + tid];
    __syncthreads();

    const float rx = 1.0f / (float)gw[0], ry = 1.0f / (float)gh[0];
    const float lx = loc[2 * l], ly = loc[2 * l + 1];

    // ---- assignment: argmin over m of (candidate ? area : inf), first-index ties
    int amin = -1; float aarea = __builtin_inff();
    for (int m = 0; m < M; ++m) {
        float cx = s_box[m][0], cy = s_box[m][1];
        float w  = s_box[m][2], h  = s_box[m][3];
        float x1 = cx - 0.5f * w, y1 = cy - 0.5f * h;
        float x2 = cx + 0.5f * w, y2 = cy + 0.5f * h;
        bool ib = (lx > x1) && (ly > y1) && (lx < x2) && (ly < y2);
        bool ic = (fabsf(lx - cx) <= rx) && (fabsf(ly - cy) <= ry);
        bool cand = s_hc[m] ? (ib && ic) : (l == s_best[m]);
        float area = w * h;
        if (cand && area < aarea) { aarea = area; amin = m; }
    }
    const bool pos = (amin >= 0);
    const int  am  = pos ? amin : 0;
    const float pf = pos ? 1.0f : 0.0f;

    const float cx = s_box[am][0], cy = s_box[am][1];
    const float bw = s_box[am][2], bh = s_box[am][3];
    const float ax1 = cx - 0.5f * bw, ay1 = cy - 0.5f * bh;
    const float ax2 = cx + 0.5f * bw, ay2 = cy + 0.5f * bh;

    const float tl = fmaxf(lx - ax1, 1e-6f), tt = fmaxf(ly - ay1, 1e-6f);
    const float tr = fmaxf(ax2 - lx, 1e-6f), tb = fmaxf(ay2 - ly, 1e-6f);
    const float hor = fminf(tl, tr) / fmaxf(fmaxf(tl, tr), 1e-6f);
    const float ver = fminf(tt, tb) / fmaxf(fmaxf(tt, tb), 1e-6f);
    const float ctr_t = sqrtf(fmaxf(hor * ver, 0.0f)) * pf;
    const float wt = pos ? fmaxf(ctr_t, 0.1f) : 0.0f;

    const float lt0 = tl * pf, lt1 = tt * pf, lt2 = tr * pf, lt3 = tb * pf;
    const float gx1 = ax1 * pf, gy1 = ay1 * pf, gx2 = ax2 * pf, gy2 = ay2 * pf;

    const int gidx = b * L + l;

    // objectness focal (numerator of mean)
    const float f_obj = focal_elem(objectness[gidx], pf);

    // smooth L1 (beta = 0.1)
    const float4 d4 = reinterpret_cast<const float4*>(box_deltas)[gidx];
    float l1 = 0.0f;
    {
        float d;
        d = fabsf(d4.x - lt0); l1 += (d < 0.1f) ? 5.0f * d * d : (d - 0.05f);
        d = fabsf(d4.y - lt1); l1 += (d < 0.1f) ? 5.0f * d * d : (d - 0.05f);
        d = fabsf(d4.z - lt2); l1 += (d < 0.1f) ? 5.0f * d * d : (d - 0.05f);
        d = fabsf(d4.w - lt3); l1 += (d < 0.1f) ? 5.0f * d * d : (d - 0.05f);
    }
    const float acc_l1 = 0.25f * l1 * wt;

    // GIoU
    const float4 p4 = reinterpret_cast<const float4*>(boxes_xyxy)[gidx];
    float iw = fmaxf(fminf(p4.z, gx2) - fmaxf(p4.x, gx1), 0.0f);
    float ih = fmaxf(fminf(p4.w, gy2) - fmaxf(p4.y, gy1), 0.0f);
    float inter = iw * ih;
    float ap = fmaxf(p4.z - p4.x, 0.0f) * fmaxf(p4.w - p4.y, 0.0f);
    float ag = fmaxf(gx2 - gx1, 0.0f) * fmaxf(gy2 - gy1, 0.0f);
    float un = ap + ag - inter;
    float iou = inter / fmaxf(un, 1e-6f);
    float hw = fmaxf(fmaxf(p4.z, gx2) - fminf(p4.x, gx1), 0.0f);
    float hh = fmaxf(fmaxf(p4.w, gy2) - fminf(p4.y, gy1), 0.0f);
    float hull = hw * hh;
    float giou = iou - (hull - un) / fmaxf(hull, 1e-6f);
    const float acc_giou = (1.0f - giou) * wt;

    // centerness BCE
    const float acc_ctr = bce_logits(centerness[gidx], ctr_t) * wt;

    s_lab[tid] = pos ? s_glab[am] : -1;
    __syncthreads();

    // ---- phase 2: coalesced class-focal sweep over the 84MB stream
    float cls_sum = 0.0f;
    const float* cbase = class_logits + (size_t)(b * L + l0) * C;
    for (int it = 0; it < C; ++it) {
        int k = it * 256 + tid;
        if ((it & 3) == 0)  // gfx1250 global_prefetch_b8, ~16KB ahead of stream
            __builtin_prefetch(cbase + k + 4096, 0, 1);
        float x = cbase[k];
        int lloc = k / C;
        int c    = k - lloc * C;
        float t = (s_lab[lloc] == c) ? 1.0f : 0.0f;
        cls_sum += focal_elem(x, t);
    }

    // ---- deterministic block reduction of 6 accumulators
    float v0 = f_obj, v1 = acc_l1, v2 = acc_giou, v3 = acc_ctr, v4 = wt, v5 = cls_sum;
    for (int off = 16; off > 0; off >>= 1) {
        v0 += __shfl_down(v0, off, 32);
        v1 += __shfl_down(v1, off, 32);
        v2 += __shfl_down(v2, off, 32);
        v3 += __shfl_down(v3, off, 32);
        v4 += __shfl_down(v4, off, 32);
        v5 += __shfl_down(v5, off, 32);
    }
    const int wave = tid >> 5, lane = tid & 31;
    if (lane == 0) {
        s_red[wave][0] = v0; s_red[wave][1] = v1; s_red[wave][2] = v2;
        s_red[wave][3] = v3; s_red[wave][4] = v4; s_red[wave][5] = v5;
    }
    __syncthreads();
    if (tid == 0) {
        float r[6] = {0, 0, 0, 0, 0, 0};
        for (int wv = 0; wv < (int)(blockDim.x >> 5); ++wv)
            for (int j = 0; j < 6; ++j) r[j] += s_red[wv][j];
        float* outp = partials + (size_t)blockIdx.x * 6;
        for (int j = 0; j < 6; ++j) outp[j] = r[j];
    }
}

// ---------------------------------------------------------------------------
// Final reduction with V_WMMA_F32_16X16X4_F32.
// B = all-ones (layout-agnostic) => D[m][n] = sum_k A[m][k] + C[m][n], columns
// identical. Per documented C/D layout, lane n holds column n with M=0..7 in
// the 8 accumulator regs (lanes 16..31 hold M=8..15). Sum of all 64 A values
// per step = (reg-sum in lane n) + (reg-sum in lane n+16).
// ---------------------------------------------------------------------------
__global__ __launch_bounds__(32)
void fcos_reduce(const float* __restrict__ partials, int nrows,
                 float inv_obj, float inv_cls, float* __restrict__ out) {
    const int lane = threadIdx.x;  // single wave32; EXEC all ones at WMMA
    float comp[6];
    const int niter = nrows >> 6;  // 64 values per WMMA step
    for (int j = 0; j < 6; ++j) {
        v8f c = {0.0f, 0.0f, 0.0f, 0.0f, 0.0f, 0.0f, 0.0f, 0.0f};
        v2f ones; ones.x = 1.0f; ones.y = 1.0f;
        for (int it = 0; it < niter; ++it) {
            int r = it * 64 + lane * 2;
            v2f a;
            a.x = partials[(size_t)r * 6 + j];
            a.y = partials[(size_t)(r + 1) * 6 + j];
            c = __builtin_amdgcn_wmma_f32_16x16x4_f32(
                    false, a, false, ones, (short)0, c, false, false);
        }
        float s = c[0] + c[1] + c[2] + c[3] + c[4] + c[5] + c[6] + c[7];
        s += __shfl_xor(s, 16, 32);
        for (int r = niter * 64; r < nrows; ++r)  // uniform tail (none for 1024)
            s += partials[(size_t)r * 6 + j];
        comp[j] = s;
    }
    if (lane == 0) {
        float wsum = comp[4];
        float loss = OBJ_W  * comp[0] * inv_obj
                   + BOX_W  * comp[1] / wsum
                   + GIOU_W * comp[2] / wsum
                   + CTR_W  * comp[3] / wsum
                   + CLS_W  * comp[5] * inv_cls;
        out[0] = loss;
    }
}

extern "C" void kernel_launch(void* const* d_in, const int* in_sizes, int n_in,
                              void* d_out, int out_size, void* d_ws, size_t ws_size,
                              hipStream_t stream) {
    const float* boxes_xyxy   = (const float*)d_in[0];
    const float* box_deltas   = (const float*)d_in[1];
    const float* class_logits = (const float*)d_in[2];
    const float* objectness   = (const float*)d_in[3];
    const float* centerness   = (const float*)d_in[4];
    const float* locations    = (const float*)d_in[5];
    const float* gt_boxes     = (const float*)d_in[6];
    const int*   gt_labels    = (const int*)d_in[7];
    const int*   grid_h       = (const int*)d_in[8];
    const int*   grid_w       = (const int*)d_in[9];

    const int L = in_sizes[5] / 2;           // locations: (L,2)
    const int B = in_sizes[3] / L;           // objectness: (B,L)
    const int M = in_sizes[7] / B;           // gt_labels: (B,M)
    const int C = in_sizes[2] / in_sizes[3]; // class_logits: (B,L,C)

    int* d_best = (int*)d_ws;                        // B*M ints
    int* d_hc   = d_best + B * M;                    // B*M ints
    size_t off  = (((size_t)2 * B * M * 4) + 255) & ~(size_t)255;
    float* d_part = (float*)((char*)d_ws + off);     // nblk*6 floats

    const int nblk = (B * L) / 256;

    fcos_assign<<<B * M, 256, 0, stream>>>(locations, gt_boxes, grid_h, grid_w,
                                           L, d_best, d_hc);
    fcos_main<<<nblk, 256, 0, stream>>>(boxes_xyxy, box_deltas, class_logits,
                                        objectness, centerness, locations,
                                        gt_boxes, gt_labels, grid_h, grid_w,
                                        d_best, d_hc, M, L, C, d_part);
    fcos_reduce<<<1, 32, 0, stream>>>(d_part, nblk,
                                      1.0f / ((float)B * (float)L),
                                      1.0f / ((float)B * (float)L * (float)C),
                                      (float*)d_out);
}